// HouseHolderFlow_5248450036640
// MI455X (gfx1250) — compile-verified
//
#include <hip/hip_runtime.h>
#include <hip/hip_bf16.h>

// HouseHolder flow via compact-WY representation.
// H_b = I - V_b * S_b * V_b^T,  V_b = [z_1 .. z_8] (128x8), S = T^T (8x8 lower-tri).
// All heavy math in FP32 WMMA (v_wmma_f32_16x16x4_f32).
// Output (268 MB, write-once, > L2) uses non-temporal stores to keep weights/h
// L2-resident.

#define BATCH   4096
#define H_DIM   512
#define Z_DIM   128
#define N_FLOW  8
#define BTILE   16          // batch rows per workgroup
#define NWAVES  8           // 8 waves * 32 lanes = 256 threads

typedef __attribute__((ext_vector_type(2))) float v2f;
typedef __attribute__((ext_vector_type(8))) float v8f;

// LDS layout (floats):
//   V   [N_FLOW][BTILE][Z_DIM]  = 16384   (z vectors for all steps)
//   Wm  [BTILE][Z_DIM][N_FLOW]  = 16384   (Wm = V * S, per element)
//   dots[BTILE][8][8]           = 1024    (z_j . z_k, j<=k; diag = norms)
//   S   [BTILE][8][8]           = 1024    (lower-triangular WY factor)
#define LDS_V    0
#define LDS_WM   16384
#define LDS_DOTS 32768
#define LDS_S    33792
#define LDS_FLOATS 34816
#define LDS_BYTES (LDS_FLOATS * 4)

__device__ __forceinline__ v8f wmma_f32_k4(v2f a, v2f b, v8f c) {
    // D = A(16x4 f32) * B(4x16 f32) + C(16x16 f32)
    return __builtin_amdgcn_wmma_f32_16x16x4_f32(
        /*neg_a=*/false, a, /*neg_b=*/false, b,
        /*c_mod=*/(short)0, c, /*reuse_a=*/false, /*reuse_b=*/false);
}

__global__ __launch_bounds__(256) void
householder_wy_kernel(const float* __restrict__ h,    // [B, H_DIM]
                      const float* __restrict__ W0,   // [Z_DIM, H_DIM]
                      const float* __restrict__ b0,   // [Z_DIM]
                      const float* __restrict__ Ws,   // [N_FLOW-1, Z_DIM, Z_DIM]
                      const float* __restrict__ bs,   // [N_FLOW-1, Z_DIM]
                      float* __restrict__ out)        // [B, Z_DIM, Z_DIM]
{
    extern __shared__ float smem[];
    float* Vs   = smem + LDS_V;     // V[s][e][z]  index: s*2048 + e*128 + z
    float* Wm   = smem + LDS_WM;    // Wm[e][n][j] index: e*1024 + n*8 + j
    float* dots = smem + LDS_DOTS;  // dots[e][j][k] index: e*64 + j*8 + k
    float* Sm   = smem + LDS_S;     // S[e][i][j]   index: e*64 + i*8 + j

    const int tid   = threadIdx.x;
    const int w     = tid >> 5;      // wave id 0..7
    const int lane  = tid & 31;
    const int lm    = lane & 15;     // A-row / B-col / C-col within fragment
    const bool hiK  = (lane >= 16);  // lanes 16-31 carry K=2,3 (A/B frags)
    const int aoff  = hiK ? 2 : 0;
    const int bBase = blockIdx.x * BTILE;

    // ---------------- Phase 1: z-chain GEMMs (WMMA f32, K=4 steps) ----------
    // Wave w owns z-columns [w*16, w*16+16). A rows = 16 batch rows of tile.
    const int zidx = w * 16 + lm;    // this lane's output column
    {
        // step 0: z1 = h @ W0^T + b0   (K = 512)
        const float* hrow  = h  + (size_t)(bBase + lm) * H_DIM;
        const float* w0row = W0 + (size_t)zidx * H_DIM;   // B[k][n] = W0[n][k]
        v8f acc = {};
        #pragma unroll 8
        for (int k0 = 0; k0 < H_DIM; k0 += 4) {
            v2f a = *(const v2f*)(hrow  + k0 + aoff);
            v2f b = *(const v2f*)(w0row + k0 + aoff);
            acc = wmma_f32_k4(a, b, acc);
        }
        const float bias = b0[zidx];
        #pragma unroll
        for (int r = 0; r < 8; ++r) {
            const int e = r + (hiK ? 8 : 0);              // C rows r / r+8
            Vs[0 * 2048 + e * 128 + zidx] = acc[r] + bias;
        }
    }
    __syncthreads();

    for (int s = 1; s < N_FLOW; ++s) {
        // z_{s+1} = z_s @ W^T + b   (K = 128)
        const float* wrow  = Ws + ((size_t)(s - 1) * Z_DIM + zidx) * Z_DIM;
        const float* aprev = Vs + (s - 1) * 2048 + lm * 128;  // A[m][k], m = lm
        v8f acc = {};
        #pragma unroll 8
        for (int k0 = 0; k0 < Z_DIM; k0 += 4) {
            v2f a = *(const v2f*)(aprev + k0 + aoff);
            v2f b = *(const v2f*)(wrow  + k0 + aoff);
            acc = wmma_f32_k4(a, b, acc);
        }
        const float bias = bs[(s - 1) * Z_DIM + zidx];
        #pragma unroll
        for (int r = 0; r < 8; ++r) {
            const int e = r + (hiK ? 8 : 0);
            Vs[s * 2048 + e * 128 + zidx] = acc[r] + bias;
        }
        __syncthreads();
    }

    // ---------------- Phase 2: pairwise dots z_j.z_k (j<=k), diag = norms ---
    for (int t = tid; t < BTILE * 36; t += 256) {
        const int e = t / 36;
        int p = t % 36, j = 0;
        while (p >= 8 - j) { p -= 8 - j; ++j; }
        const int k = j + p;
        const float* zj = Vs + j * 2048 + e * 128;
        const float* zk = Vs + k * 2048 + e * 128;
        float acc = 0.f;
        #pragma unroll 8
        for (int i = 0; i < Z_DIM; ++i) acc += zj[i] * zk[i];
        dots[e * 64 + j * 8 + k] = acc;
    }
    __syncthreads();

    // ---------------- Phase 3: triangular WY factor, S = T^T ----------------
    if (tid < BTILE) {
        const int e = tid;
        float T[8][8];
        #pragma unroll
        for (int i = 0; i < 8; ++i)
            #pragma unroll
            for (int j = 0; j < 8; ++j) T[i][j] = 0.f;
        #pragma unroll
        for (int k = 0; k < 8; ++k) {
            const float tau = 2.0f / dots[e * 64 + k * 8 + k];
            #pragma unroll
            for (int j = 0; j < 8; ++j) {
                if (j < k) {
                    float tmp = 0.f;
                    for (int i = j; i < k; ++i)
                        tmp += T[j][i] * dots[e * 64 + i * 8 + k];
                    T[j][k] = -tau * tmp;
                }
            }
            T[k][k] = tau;
        }
        #pragma unroll
        for (int i = 0; i < 8; ++i)
            #pragma unroll
            for (int j = 0; j < 8; ++j)
                Sm[e * 64 + i * 8 + j] = T[j][i];   // S = T^T (lower-tri)
    }
    __syncthreads();

    // ---------------- Phase 4: Wm = V * S  (128x8 per element) --------------
    for (int t = tid; t < BTILE * Z_DIM; t += 256) {
        const int e = t >> 7;
        const int n = t & 127;
        float v[8];
        #pragma unroll
        for (int i = 0; i < 8; ++i) v[i] = Vs[i * 2048 + e * 128 + n];
        #pragma unroll
        for (int j = 0; j < 8; ++j) {
            float acc = 0.f;
            for (int i = j; i < 8; ++i)                 // S lower-triangular
                acc += v[i] * Sm[e * 64 + i * 8 + j];
            Wm[e * 1024 + n * 8 + j] = acc;
        }
    }
    __syncthreads();

    // ---------------- Phase 5: H = I - Wm * V^T  (WMMA f32, K=8) ------------
    // 16 elems x 64 tiles = 1024 tile tasks over 8 waves (uniform per wave).
    for (int t = w; t < BTILE * 64; t += NWAVES) {
        const int e    = t >> 6;
        const int tile = t & 63;
        const int mi   = tile >> 3;
        const int ni   = tile & 7;
        const int m    = mi * 16 + lm;     // A row for this lane
        const int n    = ni * 16 + lm;     // B col for this lane

        v8f acc = {};
        #pragma unroll
        for (int k0 = 0; k0 < 8; k0 += 4) {
            v2f a = *(const v2f*)(Wm + e * 1024 + m * 8 + k0 + aoff);
            v2f b;
            b.x = Vs[(k0 + (hiK ? 2 : 0)) * 2048 + e * 128 + n];
            b.y = Vs[(k0 + (hiK ? 3 : 1)) * 2048 + e * 128 + n];
            acc = wmma_f32_k4(a, b, acc);
        }

        // Output is streamed once (write-only, > L2): non-temporal stores.
        float* outp = out + ((size_t)(bBase + e) << 14) + n;  // column fixed
        #pragma unroll
        for (int r = 0; r < 8; ++r) {
            const int mm = mi * 16 + r + (hiK ? 8 : 0);
            const float val = ((mm == n) ? 1.0f : 0.0f) - acc[r];
            __builtin_nontemporal_store(val, outp + (size_t)mm * 128);
        }
    }
}

extern "C" void kernel_launch(void* const* d_in, const int* in_sizes, int n_in,
                              void* d_out, int out_size, void* d_ws, size_t ws_size,
                              hipStream_t stream) {
    (void)in_sizes; (void)n_in; (void)d_ws; (void)ws_size; (void)out_size;
    const float* h  = (const float*)d_in[0];
    const float* W0 = (const float*)d_in[1];
    const float* b0 = (const float*)d_in[2];
    const float* Ws = (const float*)d_in[3];
    const float* bs = (const float*)d_in[4];
    float* out = (float*)d_out;

    dim3 grid(BATCH / BTILE);   // 256 workgroups
    dim3 block(256);            // 8 waves
    householder_wy_kernel<<<grid, block, LDS_BYTES, stream>>>(h, W0, b0, bs ? Ws : Ws, bs, out);
}